// NativeHGTLayer_72103910965370
// MI455X (gfx1250) — compile-verified
//
#include <hip/hip_runtime.h>
#include <stdint.h>

// ---------------------------------------------------------------------------
// HGT layer for MI455X (gfx1250).
//  - fp32 WMMA 16x16x4 for all dense GEMMs (precision-preserving; workload is
//    memory/scatter bound, so f32 matrix path is the roofline-correct pick).
//  - Weight matrices staged to LDS via the Tensor Data Mover (one 2D D#
//    descriptor, TENSORcnt) with async global->LDS copies as fallback.
//  - Edge phase reduced to gather+dot / gather+scatter by commuting the
//    per-head relation transforms to the node level.
// ---------------------------------------------------------------------------

namespace {
constexpr int NN = 30000;
constexpr int EE = 300000;
constexpr int DD = 128;   // IN == OUT
constexpr int HH = 8;
constexpr int DK = 16;
constexpr int MTILES = NN / 16;        // 1875 (exact)
constexpr int WPB = 8;                 // waves per block in GEMM kernels
constexpr size_t NF = (size_t)NN * DD;
constexpr size_t EH = (size_t)EE * HH;
}

typedef __attribute__((ext_vector_type(2))) float v2f;
typedef __attribute__((ext_vector_type(8))) float v8f;
// GCC-vector spellings matching this toolchain's builtin prototypes.
typedef int hgt_v4i __attribute__((__vector_size__(16)));
typedef unsigned int hgt_u32x4 __attribute__((__vector_size__(16)));
typedef int hgt_i32x4 __attribute__((__vector_size__(16)));
typedef int hgt_i32x8 __attribute__((__vector_size__(32)));

__device__ __forceinline__ v8f wmma_f32_k4(v2f a, v2f b, v8f c) {
  // V_WMMA_F32_16X16X4_F32: A 16x4, B 4x16, C/D 16x16 (wave32)
  return __builtin_amdgcn_wmma_f32_16x16x4_f32(
      /*neg_a=*/false, a, /*neg_b=*/false, b,
      /*c_mod=*/(short)0, c, /*reuse_a=*/false, /*reuse_b=*/false);
}

// ------------------- TDM / async global->LDS staging ------------------------
#if defined(__has_builtin)
#if __has_builtin(__builtin_amdgcn_tensor_load_to_lds)
#define HGT_TDM_BUILTIN 1
#endif
#if __has_builtin(__builtin_amdgcn_global_load_async_to_lds_b128)
#define HGT_ASYNC_BUILTIN 1
#endif
#if __has_builtin(__builtin_amdgcn_s_wait_asynccnt)
#define HGT_WAITASYNC_BUILTIN 1
#endif
#if __has_builtin(__builtin_amdgcn_s_wait_tensorcnt)
#define HGT_WAITTENSOR_BUILTIN 1
#endif
#endif

// Copy 16 bytes from global to LDS, tracked by ASYNCcnt.
// Generic LDS pointers carry the LDS byte offset in addr[31:0] (ISA 10.2),
// so integer truncation yields the correct AS(3) address value.
__device__ __forceinline__ void hgt_async_cp16(const float* gsrc, float* ldst) {
#if defined(HGT_ASYNC_BUILTIN)
  __builtin_amdgcn_global_load_async_to_lds_b128(
      (__attribute__((address_space(1))) hgt_v4i*)(unsigned long long)(uintptr_t)gsrc,
      (__attribute__((address_space(3))) hgt_v4i*)(unsigned)(uintptr_t)ldst,
      /*offset=*/0, /*cpol=*/0);
#else
  asm volatile("global_load_async_to_lds_b128 %0, %1, off"
               :: "v"((unsigned)(uintptr_t)ldst),
                  "v"((unsigned long long)(uintptr_t)gsrc)
               : "memory");
#endif
}

__device__ __forceinline__ void hgt_wait_async0() {
#if defined(HGT_WAITASYNC_BUILTIN)
  __builtin_amdgcn_s_wait_asynccnt(0);
#else
  asm volatile("s_wait_asynccnt 0x0" ::: "memory");
#endif
}

__device__ __forceinline__ void hgt_wait_tensor0() {
#if defined(HGT_WAITTENSOR_BUILTIN)
  __builtin_amdgcn_s_wait_tensorcnt(0);
#else
  asm volatile("s_wait_tensorcnt 0x0" ::: "memory");
#endif
}

// Stage one 128x128 f32 matrix (64 KB, contiguous row-major) into LDS.
// Called by all 256 threads of the block; ends fully published (barrier).
__device__ __forceinline__ void hgt_stage_matrix(const float* __restrict__ W,
                                                 float* __restrict__ sW, int tid) {
#if defined(HGT_TDM_BUILTIN)
  if ((tid >> 5) == 0) {   // wave 0 issues the TDM descriptor (EXEC ignored)
    const unsigned long long ga = (unsigned long long)(uintptr_t)W;
    const unsigned lds = (unsigned)(uintptr_t)sW;     // LDS byte offset
    // D# group 0: count=1 | lds_addr | global_addr[56:0] | type=2 ("image")
    hgt_u32x4 g0 = { 1u, lds, (unsigned)(ga & 0xffffffffull),
                     (unsigned)((ga >> 32) & 0x01ffffffull) | 0x80000000u };
    // D# group 1: data_size=4B; tensor_dim0=128, tensor_dim1=128;
    //             tile_dim0=128, tile_dim1=128, tile_dim2=0; dim0_stride=128
    hgt_i32x8 g1 = { (int)(2u << 16),        // data_size code 2 (4 bytes)
                     (int)(128u << 16),      // tensor_dim0[15:0] in [63:48]
                     (int)(128u << 16),      // tensor_dim1[15:0] in [95:80]
                     (int)(128u << 16),      // tile_dim0 in [127:112]
                     128,                    // tile_dim1 in [143:128]
                     128,                    // tensor_dim0_stride[31:0]
                     0, 0 };
    hgt_i32x4 gz4 = { 0, 0, 0, 0 };          // groups 2/3 unused (<=2D)
    hgt_i32x8 gz8 = { 0, 0, 0, 0, 0, 0, 0, 0 };
    __builtin_amdgcn_tensor_load_to_lds(g0, g1, gz4, gz4, gz8, /*cpol=*/0);
  }
  hgt_wait_tensor0();       // wave 0 waits its DMA; others pass (TENSORcnt==0)
  __syncthreads();
#else
  // Fallback: per-lane async b128 copies (ASYNCcnt path).
#pragma unroll
  for (int it = 0; it < (DD * DD / 4) / 256; ++it) {
    const int g4 = it * 256 + tid;            // 16-byte group index
    hgt_async_cp16(W + (size_t)g4 * 4, &sW[g4 * 4]);
  }
  hgt_wait_async0();
  __syncthreads();
#endif
}

// Ordered-uint encoding for float atomic max (memset-0 == -inf identity).
__device__ __forceinline__ unsigned hgt_ford(float f) {
  unsigned u = __float_as_uint(f);
  return (u & 0x80000000u) ? ~u : (u | 0x80000000u);
}
__device__ __forceinline__ float hgt_forddec(unsigned u) {
  return __uint_as_float((u & 0x80000000u) ? (u & 0x7fffffffu) : ~u);
}

// ----------------------------------------------------------------------------
// Dense [rows x 128] @ [128 x 128] + bias (optional gated-skip blend).
// 8 waves/block; W staged once into LDS (64 KB, TDM); each wave computes one
// 16-row M-tile with f32 WMMA, B fragments served from LDS.
// ----------------------------------------------------------------------------
__global__ __launch_bounds__(256) void hgt_gemm128_lds(
    const float* __restrict__ A, const float* __restrict__ W,
    const float* __restrict__ bias, float* __restrict__ Out,
    const float* __restrict__ Hin, const float* __restrict__ skip) {
  __shared__ float sW[DD * DD];   // 64 KB, row-major copy of W

  const int tid = threadIdx.x;
  hgt_stage_matrix(W, sW, tid);

  const int lane = tid & 31;
  const int wave = tid >> 5;
  int tile = blockIdx.x * WPB + wave;
  const bool valid = tile < MTILES;
  if (!valid) tile = MTILES - 1;               // clamp: EXEC stays full for WMMA
  const int m0 = tile * 16;
  const int row = m0 + (lane & 15);
  const int kb2 = (lane >> 4) << 1;            // K 0/1 vs 2/3 per half-wave
  const int cl = lane & 15;
  const int rh = (lane >> 4) << 3;             // D-matrix row half
  const float* arow = A + (size_t)row * DD;

  v2f afr[32];
#pragma unroll
  for (int kk = 0; kk < 32; ++kk) {
    const int k0 = (kk << 2) + kb2;
    afr[kk].x = arow[k0];
    afr[kk].y = arow[k0 + 1];
  }

  float alpha = 1.0f, beta = 0.0f;
  if (skip) { alpha = 1.0f / (1.0f + __expf(-skip[0])); beta = 1.0f - alpha; }

  for (int nt = 0; nt < 8; ++nt) {
    const int col = (nt << 4) + cl;
    v8f acc = {};
#pragma unroll
    for (int kk = 0; kk < 32; ++kk) {
      const int k0 = (kk << 2) + kb2;
      v2f bfr;
      bfr.x = sW[k0 * DD + col];
      bfr.y = sW[(k0 + 1) * DD + col];
      acc = wmma_f32_k4(afr[kk], bfr, acc);
    }
    if (valid) {                                // wave-uniform store guard
      const float bv = bias[col];
#pragma unroll
      for (int g = 0; g < 8; ++g) {
        const size_t idx = (size_t)(m0 + rh + g) * DD + col;
        const float r = acc[g] + bv;
        Out[idx] = Hin ? (r * alpha + Hin[idx] * beta) : r;
      }
    }
  }
}

// ----------------------------------------------------------------------------
// Per-head 16x16 transform at node level: Out[n, h*16+c] = sum_k X[n,h*16+k]*B[k][c]
// transposeB=1: B[k][c] = R_h[c*16+k]  (qt = q @ rel_attT, scaled by pri/sqrt(dk))
// transposeB=0: B[k][c] = R_h[k*16+c]  (t  = s @ rel_msg)
// ----------------------------------------------------------------------------
__global__ void hgt_headgemm(const float* __restrict__ X, const float* __restrict__ R,
                             const float* __restrict__ pri, float* __restrict__ Out,
                             int transposeB) {
  const int lane = threadIdx.x;
  const int m0 = blockIdx.x * 16;
  const int row = m0 + (lane & 15);
  const int kb2 = (lane >> 4) << 1;
  const int cl = lane & 15;
  const int rh = (lane >> 4) << 3;

  for (int h = 0; h < HH; ++h) {
    const float* xrow = X + (size_t)row * DD + h * DK;
    const float* Rh = R + h * DK * DK;
    v8f acc = {};
#pragma unroll
    for (int kk = 0; kk < 4; ++kk) {
      const int k0 = (kk << 2) + kb2;
      v2f a; a.x = xrow[k0]; a.y = xrow[k0 + 1];
      v2f b;
      if (transposeB) { b.x = Rh[cl * DK + k0];  b.y = Rh[cl * DK + k0 + 1]; }
      else            { b.x = Rh[k0 * DK + cl];  b.y = Rh[(k0 + 1) * DK + cl]; }
      acc = wmma_f32_k4(a, b, acc);
    }
    const float sc = pri ? (pri[h] * 0.25f) : 1.0f;   // 1/sqrt(16) = 0.25
#pragma unroll
    for (int g = 0; g < 8; ++g)
      Out[(size_t)(m0 + rh + g) * DD + h * DK + cl] = acc[g] * sc;
  }
}

// ----------------------------------------------------------------------------
// Edge phase. One thread per (edge, head).
// ----------------------------------------------------------------------------
__global__ void hgt_edge_att(const float* __restrict__ kp, const float* __restrict__ qt,
                             const int* __restrict__ src, const int* __restrict__ dst,
                             float* __restrict__ attb, unsigned* __restrict__ mord, int nE) {
  const int t = blockIdx.x * blockDim.x + threadIdx.x;
  if (t >= nE * HH) return;
  const int e = t >> 3, h = t & 7;
  const int s = src[e], d = dst[e];
  const float4* a = (const float4*)(kp + (size_t)s * DD + h * DK);
  const float4* b = (const float4*)(qt + (size_t)d * DD + h * DK);
  float acc = 0.f;
#pragma unroll
  for (int i = 0; i < 4; ++i) {
    float4 x = a[i], y = b[i];
    acc += x.x * y.x + x.y * y.y + x.z * y.z + x.w * y.w;
  }
  attb[t] = acc;   // t == e*8 + h
  atomicMax(mord + (size_t)d * HH + h, hgt_ford(acc));
}

__global__ void hgt_edge_exp(float* __restrict__ attb, const int* __restrict__ dst,
                             const unsigned* __restrict__ mord, float* __restrict__ z, int nE) {
  const int t = blockIdx.x * blockDim.x + threadIdx.x;
  if (t >= nE * HH) return;
  const int e = t >> 3, h = t & 7;
  const int d = dst[e];
  const float m = hgt_forddec(mord[(size_t)d * HH + h]);
  const float w = __expf(attb[t] - m);
  attb[t] = w;
  atomicAdd(z + (size_t)d * HH + h, w);
}

__global__ void hgt_edge_agg(const float* __restrict__ attb, const int* __restrict__ src,
                             const int* __restrict__ dst, const float* __restrict__ z,
                             const float* __restrict__ vp, float* __restrict__ sacc, int nE) {
  const int t = blockIdx.x * blockDim.x + threadIdx.x;
  if (t >= nE * HH) return;
  const int e = t >> 3, h = t & 7;
  const int s = src[e], d = dst[e];
  const float c = attb[t] / z[(size_t)d * HH + h];
  const float* v = vp + (size_t)s * DD + h * DK;
  float* o = sacc + (size_t)d * DD + h * DK;
#pragma unroll
  for (int i = 0; i < DK; ++i) atomicAdd(o + i, c * v[i]);
}

// ----------------------------------------------------------------------------
extern "C" void kernel_launch(void* const* d_in, const int* in_sizes, int n_in,
                              void* d_out, int out_size, void* d_ws, size_t ws_size,
                              hipStream_t stream) {
  (void)in_sizes; (void)n_in; (void)out_size; (void)ws_size;
  const float* h0 = (const float*)d_in[0];
  const float* h1 = (const float*)d_in[1];
  const int* src0 = (const int*)d_in[2];
  const int* dst0 = (const int*)d_in[3];
  const int* src1 = (const int*)d_in[4];
  const int* dst1 = (const int*)d_in[5];
  const float* kw = (const float*)d_in[6];
  const float* kbias = (const float*)d_in[7];
  const float* qw = (const float*)d_in[8];
  const float* qbias = (const float*)d_in[9];
  const float* vw = (const float*)d_in[10];
  const float* vbias = (const float*)d_in[11];
  const float* aw = (const float*)d_in[12];
  const float* ab = (const float*)d_in[13];
  const float* rel_att = (const float*)d_in[14];
  const float* rel_msg = (const float*)d_in[15];
  const float* rel_pri = (const float*)d_in[16];
  const float* skip = (const float*)d_in[17];
  float* out = (float*)d_out;

  float* ws = (float*)d_ws;
  size_t o = 0;
  auto take = [&](size_t n) { float* p = ws + o; o += n; return p; };
  float* kp0 = take(NF); float* vp0 = take(NF); float* qp0 = take(NF);
  float* kp1 = take(NF); float* vp1 = take(NF); float* qp1 = take(NF);
  float* qt0 = take(NF); float* qt1 = take(NF);
  float* tb0 = take(NF); float* tb1 = take(NF);
  float* att0 = take(EH); float* att1 = take(EH);
  // contiguous zero-init region: s0, s1, z0, z1, m0, m1
  float* s0 = take(NF); float* s1 = take(NF);
  float* z0 = take((size_t)NN * HH); float* z1 = take((size_t)NN * HH);
  unsigned* mm0 = (unsigned*)take((size_t)NN * HH);
  unsigned* mm1 = (unsigned*)take((size_t)NN * HH);
  const size_t zero_bytes = (2 * NF + 4 * (size_t)NN * HH) * sizeof(float);
  (void)hipMemsetAsync(s0, 0, zero_bytes, stream);

  const dim3 gblk(256);                          // 8 waves share staged W
  const dim3 gridG((MTILES + WPB - 1) / WPB);    // 235 blocks
  const dim3 wblk(32);
  const dim3 gridM(MTILES);                      // 1875 single-wave blocks
  const dim3 eblk(256);
  const dim3 egrd((EE * HH + 255) / 256);

  // 1) K/V/Q projections (WMMA f32, TDM-staged weights)
  hgt_gemm128_lds<<<gridG, gblk, 0, stream>>>(h0, kw + 0 * DD * DD, kbias + 0 * DD, kp0, nullptr, nullptr);
  hgt_gemm128_lds<<<gridG, gblk, 0, stream>>>(h0, vw + 0 * DD * DD, vbias + 0 * DD, vp0, nullptr, nullptr);
  hgt_gemm128_lds<<<gridG, gblk, 0, stream>>>(h0, qw + 0 * DD * DD, qbias + 0 * DD, qp0, nullptr, nullptr);
  hgt_gemm128_lds<<<gridG, gblk, 0, stream>>>(h1, kw + 1 * DD * DD, kbias + 1 * DD, kp1, nullptr, nullptr);
  hgt_gemm128_lds<<<gridG, gblk, 0, stream>>>(h1, vw + 1 * DD * DD, vbias + 1 * DD, vp1, nullptr, nullptr);
  hgt_gemm128_lds<<<gridG, gblk, 0, stream>>>(h1, qw + 1 * DD * DD, qbias + 1 * DD, qp1, nullptr, nullptr);

  // 2) qt = q @ rel_att^T * pri/sqrt(dk)  (node-level, WMMA)
  hgt_headgemm<<<gridM, wblk, 0, stream>>>(qp1, rel_att + 0 * HH * DK * DK, rel_pri + 0 * HH, qt0, 1);
  hgt_headgemm<<<gridM, wblk, 0, stream>>>(qp0, rel_att + 1 * HH * DK * DK, rel_pri + 1 * HH, qt1, 1);

  // 3) edge attention scores + segment max
  hgt_edge_att<<<egrd, eblk, 0, stream>>>(kp0, qt0, src0, dst0, att0, mm0, EE);
  hgt_edge_att<<<egrd, eblk, 0, stream>>>(kp1, qt1, src1, dst1, att1, mm1, EE);

  // 4) exp(att - max) + segment sum
  hgt_edge_exp<<<egrd, eblk, 0, stream>>>(att0, dst0, mm0, z0, EE);
  hgt_edge_exp<<<egrd, eblk, 0, stream>>>(att1, dst1, mm1, z1, EE);

  // 5) scatter-aggregate s[dst] += softmax * v[src]
  hgt_edge_agg<<<egrd, eblk, 0, stream>>>(att0, src0, dst0, z0, vp0, s0, EE);
  hgt_edge_agg<<<egrd, eblk, 0, stream>>>(att1, src1, dst1, z1, vp1, s1, EE);

  // 6) t = s @ rel_msg (node-level, WMMA)
  hgt_headgemm<<<gridM, wblk, 0, stream>>>(s0, rel_msg + 0 * HH * DK * DK, nullptr, tb1, 0);
  hgt_headgemm<<<gridM, wblk, 0, stream>>>(s1, rel_msg + 1 * HH * DK * DK, nullptr, tb0, 0);

  // 7) output projection + gated skip (WMMA, TDM-staged weights)
  hgt_gemm128_lds<<<gridG, gblk, 0, stream>>>(tb0, aw + 0 * DD * DD, ab + 0 * DD, out, h0, skip + 0);
  hgt_gemm128_lds<<<gridG, gblk, 0, stream>>>(tb1, aw + 1 * DD * DD, ab + 1 * DD, out + NF, h1, skip + 1);
}